// CosineSimilarity_14534169330171
// MI455X (gfx1250) — compile-verified
//
#include <hip/hip_runtime.h>

// Problem constants (from reference):
//   C=8 refs, T=128 targets, P=1024 positions (32x32 grid), D=256, out 64x64.
#define NC   8
#define NT   128
#define NP   1024
#define ND   256
#define GRID_IN  32
#define GRID_OUT 64

typedef __attribute__((ext_vector_type(2))) float v2f;
typedef __attribute__((ext_vector_type(8))) float v8f;

// ---------------------------------------------------------------------------
// Kernel 1: fused normalize + (ref x target^T) GEMM via V_WMMA_F32_16X16X4_F32.
// One wave (32 lanes) computes a 16-row x 16-col (8 valid) tile of sim.
// Block = 256 threads = 8 waves = 8 row-tiles.  Grid = 131072/16/8 = 1024.
// ---------------------------------------------------------------------------
__global__ void __launch_bounds__(256)
sim_wmma_kernel(const float* __restrict__ refg,   // [NC][ND]
                const float* __restrict__ tgt,    // [NT*NP][ND]
                float* __restrict__ sim)          // [NT][NC][NP]
{
    // Stage B matrix (ref, transposed to [k][n], cols 8..15 zero) in LDS: 16KB.
    __shared__ float ldsB[ND * 16];
    {
        const int k = threadIdx.x;            // 0..255 == one k per thread
        #pragma unroll
        for (int n = 0; n < 16; ++n)
            ldsB[k * 16 + n] = (n < NC) ? refg[n * ND + k] : 0.0f;
    }
    __syncthreads();

    const int lane   = threadIdx.x & 31;
    const int waveId = threadIdx.x >> 5;
    const int m      = lane & 15;     // row within tile (A-matrix M)
    const int h      = lane >> 4;     // half-wave selector (K sub-pair)
    const int n      = lane & 15;     // column (B-matrix N)

    const int tile   = blockIdx.x * 8 + waveId;       // 16-row tile index
    const int rowBase = tile * 16;

    // Lane's A-source pointer: row (rowBase+m), starting at column 2*h.
    const float* __restrict__ rowptr =
        tgt + (size_t)(rowBase + m) * ND + 2 * h;

    v8f   acc   = {};
    float sumsq = 0.0f;

    #pragma unroll 8
    for (int k = 0; k < ND; k += 4) {
        // A fragment (f32 16x4): lane holds elements {k+2h, k+2h+1} of its row.
        const float2 av = *reinterpret_cast<const float2*>(rowptr + k);
        sumsq += av.x * av.x + av.y * av.y;
        v2f a; a[0] = av.x; a[1] = av.y;

        // B fragment (f32 4x16): lane holds B[k+2h][n], B[k+2h+1][n] from LDS.
        v2f b;
        b[0] = ldsB[(k + 2 * h)     * 16 + n];
        b[1] = ldsB[(k + 2 * h + 1) * 16 + n];

        // D = A x B + C   (8 args: neg_a, A, neg_b, B, c_mod, C, reuse_a, reuse_b)
        acc = __builtin_amdgcn_wmma_f32_16x16x4_f32(
                  false, a, false, b, (short)0, acc, false, false);
    }

    // Full row sum-of-squares: lane m (h=0) + lane m+16 (h=1).
    sumsq += __shfl_xor(sumsq, 16, 32);
    const float inv = rsqrtf(sumsq);          // lane L holds 1/||row (L&15)||

    // C/D layout: VGPR v, half h -> row (v + 8h); column = lane & 15.
    #pragma unroll
    for (int v = 0; v < 8; ++v) {
        const int mo = v + 8 * h;
        const float invRow = __shfl(inv, mo, 32);   // norm of row mo
        if (n < NC) {
            const int r = rowBase + mo;             // global (t,p) row
            const int t = r >> 10;                  // / NP
            const int p = r & (NP - 1);
            sim[((size_t)t * NC + n) * NP + p] = acc[v] * invRow;
        }
    }
}

// ---------------------------------------------------------------------------
// Kernel 2: separable bilinear 32x32 -> 64x64 (JAX half-pixel convention).
// in = o/2 - 0.25  =>  even o=2k: (k-1,0.25),(k,0.75); odd o=2k+1: (k,0.75),(k+1,0.25)
// Edge clamp == JAX weight renormalization.  One block per (t,c) image.
// ---------------------------------------------------------------------------
__device__ __forceinline__ void bil_taps(int o, int& lo, int& hi,
                                         float& wlo, float& whi)
{
    const int k = o >> 1;
    if (o & 1) { lo = k;                hi = min(k + 1, GRID_IN - 1); wlo = 0.75f; whi = 0.25f; }
    else       { lo = max(k - 1, 0);    hi = k;                       wlo = 0.25f; whi = 0.75f; }
}

__global__ void __launch_bounds__(256)
resize_kernel(const float* __restrict__ sim,   // [NT*NC][1024]
              float* __restrict__ out)         // [NT*NC][64*64]
{
    __shared__ float tile[GRID_IN * GRID_IN];
    const float* __restrict__ src = sim + (size_t)blockIdx.x * (GRID_IN * GRID_IN);
    for (int i = threadIdx.x; i < GRID_IN * GRID_IN; i += 256)
        tile[i] = src[i];
    __syncthreads();

    float* __restrict__ dst = out + (size_t)blockIdx.x * (GRID_OUT * GRID_OUT);
    for (int idx = threadIdx.x; idx < GRID_OUT * GRID_OUT; idx += 256) {
        const int ox = idx & (GRID_OUT - 1);
        const int oy = idx >> 6;
        int xlo, xhi, ylo, yhi; float wxlo, wxhi, wylo, wyhi;
        bil_taps(ox, xlo, xhi, wxlo, wxhi);
        bil_taps(oy, ylo, yhi, wylo, wyhi);
        const float r0 = wxlo * tile[ylo * GRID_IN + xlo] + wxhi * tile[ylo * GRID_IN + xhi];
        const float r1 = wxlo * tile[yhi * GRID_IN + xlo] + wxhi * tile[yhi * GRID_IN + xhi];
        dst[idx] = wylo * r0 + wyhi * r1;
    }
}

// ---------------------------------------------------------------------------
extern "C" void kernel_launch(void* const* d_in, const int* in_sizes, int n_in,
                              void* d_out, int out_size, void* d_ws, size_t ws_size,
                              hipStream_t stream)
{
    (void)in_sizes; (void)n_in; (void)out_size; (void)ws_size;
    const float* refg = (const float*)d_in[0];   // (8,1,256) f32
    const float* tgt  = (const float*)d_in[1];   // (128,1024,256) f32
    // d_in[2] (category_ids == arange(8)) is mathematically unused.

    float* sim = (float*)d_ws;                   // [128][8][1024] = 4 MB scratch

    // 131072 rows / 16 per wave / 8 waves per block = 1024 blocks
    sim_wmma_kernel<<<1024, 256, 0, stream>>>(refg, tgt, sim);

    // One block per (t,c) image: 128*8 = 1024 blocks
    resize_kernel<<<NT * NC, 256, 0, stream>>>(sim, (float*)d_out);
}